// KalmanFilter_80341658239324
// MI455X (gfx1250) — compile-verified
//
#include <hip/hip_runtime.h>
#include <stdint.h>

#define GDIM 128
#define TDIM 512
#define SDIM 24
#define MDIM 4
#define PSTR 33  // padded LDS row stride for 32x32 P / T1 (col 24 carries the mean)
#define QSTR 32  // padded row stride for Q in LDS
#define HPSTR 32 // padded row stride for HP (16x32; rows 4..15 zero; col 24 carries mu)

typedef __attribute__((ext_vector_type(2))) float v2f;
typedef __attribute__((ext_vector_type(8))) float v8f;

// ---------------- CDNA5 async copy to LDS (ASYNCcnt path) ----------------

__device__ __forceinline__ unsigned lds_off_u32(const void* p) {
  return (unsigned)(unsigned long long)p;  // low 32 bits = LDS byte offset
}

__device__ __forceinline__ void async_ld_b128(unsigned lds_byte, const void* gbase,
                                              unsigned gbyte) {
  asm volatile("global_load_async_to_lds_b128 %0, %1, %2"
               :: "v"(lds_byte), "v"(gbyte), "s"(gbase)
               : "memory");
}

__device__ __forceinline__ void wait_async0() {
  asm volatile("s_wait_asynccnt 0" ::: "memory");
}

// ---------------- WMMA f32 16x16x4 fragment helpers (all unconditional) ----------------
// A 16x4 (MxK): lanes 0-15 hold M=lane, K={k0,k0+1}; lanes 16-31 hold M=lane-16, K={k0+2,k0+3}
// B 4x16 (KxN): lanes 0-15 hold N=lane, K={k0,k0+1};  lanes 16-31 hold N=lane-16, K={k0+2,k0+3}
// C/D 16x16:    VGPR v: lanes 0-15 -> (row0+v, col0+lane); lanes 16-31 -> (row0+8+v, col0+lane-16)

__device__ __forceinline__ v2f wmma_load_a(const float* src, int stride,
                                           int row0, int k0, int lane) {
  int r = row0 + (lane & 15);
  int k = k0 + ((lane >> 4) << 1);
  v2f a;
  a.x = src[r * stride + k];
  a.y = src[r * stride + k + 1];
  return a;
}

__device__ __forceinline__ v2f wmma_load_b(const float* src, int stride,
                                           int k0, int col0, int lane) {
  int n = col0 + (lane & 15);
  int k = k0 + ((lane >> 4) << 1);
  v2f b;
  b.x = src[k * stride + n];
  b.y = src[(k + 1) * stride + n];
  return b;
}

// B tile of X^T: B[k][n] = X[n*stride + k]; source must have >= col0+16 zero-padded rows
__device__ __forceinline__ v2f wmma_load_bT(const float* src, int stride,
                                            int k0, int col0, int lane) {
  int n = col0 + (lane & 15);
  int k = k0 + ((lane >> 4) << 1);
  v2f b;
  b.x = src[n * stride + k];
  b.y = src[n * stride + k + 1];
  return b;
}

__device__ __forceinline__ v8f wmma_load_c(const float* src, int stride, int row0, int col0,
                                           int lane) {
  int c = col0 + (lane & 15);
  int rb = row0 + ((lane >> 4) << 3);
  v8f r;
#pragma unroll
  for (int v = 0; v < 8; ++v) r[v] = src[(rb + v) * stride + c];
  return r;
}

__device__ __forceinline__ void wmma_store_d(float* dst, int stride, int row0, int col0,
                                             v8f d, int lane) {
  int c = col0 + (lane & 15);
  int rb = row0 + ((lane >> 4) << 3);
#pragma unroll
  for (int v = 0; v < 8; ++v) dst[(rb + v) * stride + c] = d[v];
}

// Q buffer is 32x32 zero-padded: unconditional read
__device__ __forceinline__ void wmma_store_d_plus_q(float* dst, int stride, int row0, int col0,
                                                    v8f d, const float* Qs, int lane) {
  int c = col0 + (lane & 15);
  int rb = row0 + ((lane >> 4) << 3);
#pragma unroll
  for (int v = 0; v < 8; ++v) {
    int r = rb + v;
    dst[r * stride + c] = d[v] + Qs[r * QSTR + c];
  }
}

#define WMMA_F32(A, B, C) \
  __builtin_amdgcn_wmma_f32_16x16x4_f32(false, (A), false, (B), (short)0, (C), false, false)

// ---------------- per-step input staging (double buffered) ----------------

__device__ __forceinline__ void stage_inputs(int lane, int gt,
                                             const float* yin, const float* Fin,
                                             const float* Qin, const float* Hin,
                                             const float* Rin,
                                             float* dF, float* dQ, float* dH,
                                             float* dR, float* dY) {
  const float* Fp = Fin + (size_t)gt * (SDIM * SDIM);
  const float* Qp = Qin + (size_t)gt * (SDIM * SDIM);
  const float* Hp = Hin + (size_t)gt * (MDIM * SDIM);
  const float* Rp = Rin + (size_t)gt * (MDIM * MDIM);
  const float* Yp = yin + (size_t)gt * MDIM;
  unsigned oF = lds_off_u32(dF), oQ = lds_off_u32(dQ);
  unsigned oH = lds_off_u32(dH), oR = lds_off_u32(dR), oY = lds_off_u32(dY);
  for (int c = lane; c < (SDIM * SDIM) / 4; c += 32) {  // 144 float4 chunks, 6 per 24-row
    async_ld_b128(oF + (unsigned)c * 16u, Fp, (unsigned)c * 16u);
    unsigned qr = (unsigned)c / 6u, qc = (unsigned)c % 6u;   // Q into stride-32 rows
    async_ld_b128(oQ + qr * (QSTR * 4u) + qc * 16u, Qp, (unsigned)c * 16u);
  }
  if (lane < (MDIM * SDIM) / 4) async_ld_b128(oH + (unsigned)lane * 16u, Hp, (unsigned)lane * 16u);
  if (lane < (MDIM * MDIM) / 4) async_ld_b128(oR + (unsigned)lane * 16u, Rp, (unsigned)lane * 16u);
  if (lane == 0) async_ld_b128(oY, Yp, 0u);
}

// ---------------- kernel: one wave32 per Kalman chain ----------------

__global__ void __launch_bounds__(32)
kalman_wmma_kernel(const float* __restrict__ yin, const float* __restrict__ Fin,
                   const float* __restrict__ Qin, const float* __restrict__ Hin,
                   const float* __restrict__ Rin, const float* __restrict__ m0in,
                   const float* __restrict__ P0in, float* __restrict__ out) {
  __shared__ __align__(16) float Pm[32 * PSTR];      // [P | m] zero-padded to 32x32
  __shared__ __align__(16) float Tm[32 * PSTR];      // temp T1 = F * [P_u | m_u]
  __shared__ __align__(16) float bF[2][32 * SDIM];   // F: 24x24 + rows 24..31 zero
  __shared__ __align__(16) float bQ[2][32 * QSTR];   // Q: 24x24 in stride-32 rows, pad zero
  __shared__ __align__(16) float bH[2][16 * SDIM];   // H: 4x24 + rows 4..15 zero
  __shared__ __align__(16) float bR[2][MDIM * MDIM];
  __shared__ __align__(16) float bY[2][4];
  __shared__ __align__(16) float HPs[16 * HPSTR];    // [HP | mu], rows 4..15 permanently zero
  __shared__ __align__(16) float nKt[32 * MDIM];     // -Kt^T (24x4) + rows 24..31 zero
  __shared__ float Smt[MDIM * MDIM];                 // Sig - R (pre-R innovation cov)

  const int lane = threadIdx.x;
  const int g = blockIdx.x;

  float* means = out;                                 // (G,T,M)
  float* covs = out + (size_t)GDIM * TDIM * MDIM;     // (G,T,M,M)

  // kick off staging of t=0 inputs immediately (HBM round trip overlaps init)
  stage_inputs(lane, g * TDIM + 0, yin, Fin, Qin, Hin, Rin,
               bF[0], bQ[0], bH[0], bR[0], bY[0]);

  // one-time zero of WMMA padding regions (disjoint from async-written cells)
  for (int i = lane; i < 2 * 8 * SDIM; i += 32) {     // bF rows 24..31
    int bsel = i / (8 * SDIM);
    bF[bsel][SDIM * SDIM + (i % (8 * SDIM))] = 0.0f;
  }
  for (int i = lane; i < 2 * 12 * SDIM; i += 32) {    // bH rows 4..15
    int bsel = i / (12 * SDIM);
    bH[bsel][MDIM * SDIM + (i % (12 * SDIM))] = 0.0f;
  }
  for (int i = lane; i < 2 * 32 * QSTR; i += 32) {    // bQ pad cells
    int bsel = i >> 10;
    int idx = i & 1023;
    int r = idx >> 5, c = idx & 31;
    if (r >= SDIM || c >= SDIM) bQ[bsel][idx] = 0.0f;
  }
  for (int i = lane; i < 12 * HPSTR; i += 32)         // HPs rows 4..15 (never rewritten)
    HPs[MDIM * HPSTR + i] = 0.0f;
  nKt[SDIM * MDIM + lane] = 0.0f;                     // rows 24..31

  // init state: P zero-padded 32x32 (col 24 will carry m), mean in register
  for (int i = lane; i < 32 * PSTR; i += 32) Pm[i] = 0.0f;
  __syncthreads();
  for (int e = lane; e < SDIM * SDIM; e += 32)
    Pm[(e / SDIM) * PSTR + (e % SDIM)] = P0in[(size_t)g * SDIM * SDIM + e];
  float mreg = m0in[(size_t)g * SDIM + ((lane < SDIM) ? lane : 0)];

  wait_async0();
  __syncthreads();

  for (int t = 0; t < TDIM; ++t) {
    const int b = t & 1;
    if (t + 1 < TDIM)
      stage_inputs(lane, g * TDIM + t + 1, yin, Fin, Qin, Hin, Rin,
                   bF[b ^ 1], bQ[b ^ 1], bH[b ^ 1], bR[b ^ 1], bY[b ^ 1]);

    // deposit current mean into column 24 of Pm
    if (lane < SDIM) Pm[lane * PSTR + 24] = mreg;
    __syncthreads();

    const float* Fs = bF[b];
    const float* Qs = bQ[b];
    const float* Hs = bH[b];
    const float* Rs = bR[b];
    const float* Ys = bY[b];

    // ---- [HP | mu] = Hpad (16x24) x [P | m] via WMMA (12 ops) ----
    {
      v8f h0 = {}, h1 = {};
#pragma unroll
      for (int k0 = 0; k0 < SDIM; k0 += 4) {
        v2f a = wmma_load_a(Hs, SDIM, 0, k0, lane);
        v2f b0 = wmma_load_b(Pm, PSTR, k0, 0, lane);
        v2f b1 = wmma_load_b(Pm, PSTR, k0, 16, lane);
        h0 = WMMA_F32(a, b0, h0);
        h1 = WMMA_F32(a, b1, h1);
      }
      if (lane < 16) {
#pragma unroll
        for (int v = 0; v < MDIM; ++v) {
          HPs[v * HPSTR + lane] = h0[v];
          HPs[v * HPSTR + 16 + lane] = h1[v];
        }
      }
    }
    __syncthreads();

    // ---- Sig - R = HP * H^T via WMMA (6 ops; rows/cols 0..3 valid) ----
    {
      v8f sg = {};
#pragma unroll
      for (int k0 = 0; k0 < SDIM; k0 += 4) {
        v2f a = wmma_load_a(HPs, HPSTR, 0, k0, lane);     // HP rows 0..3 (+zero pad)
        v2f bb = wmma_load_bT(Hs, SDIM, k0, 0, lane);     // H^T (pad rows of H are zero)
        sg = WMMA_F32(a, bb, sg);
      }
      if (lane < MDIM) {
#pragma unroll
        for (int v = 0; v < MDIM; ++v) Smt[v * MDIM + lane] = sg[v];
      }
    }
    __syncthreads();

    // ---- emit outputs for time t ----
    {
      float muv = HPs[(lane & 3) * HPSTR + 24];              // mu from the HP WMMA
      float sgv = Smt[lane & 15] + Rs[lane & 15];            // Sig = HP*H^T + R
      if (lane < MDIM) means[((size_t)g * TDIM + t) * MDIM + lane] = muv;
      if (lane < 16) covs[((size_t)g * TDIM + t) * 16 + lane] = sgv;
    }

    if (t + 1 < TDIM) {
      // ---- 4x4 inverse of Sm: redundant per-lane Gauss-Jordan (SPD, no pivot) ----
      float A_[16], I_[16];
#pragma unroll
      for (int i = 0; i < 16; ++i) { A_[i] = Smt[i] + Rs[i]; I_[i] = 0.0f; }
#pragma unroll
      for (int i = 0; i < 4; ++i) I_[i * 4 + i] = 1.0f;
#pragma unroll
      for (int kk = 0; kk < 4; ++kk) {
        float p = 1.0f / A_[kk * 4 + kk];
#pragma unroll
        for (int j = 0; j < 4; ++j) { A_[kk * 4 + j] *= p; I_[kk * 4 + j] *= p; }
#pragma unroll
        for (int i = 0; i < 4; ++i) {
          if (i == kk) continue;
          float f = A_[i * 4 + kk];
#pragma unroll
          for (int j = 0; j < 4; ++j) {
            A_[i * 4 + j] -= f * A_[kk * 4 + j];
            I_[i * 4 + j] -= f * I_[kk * 4 + j];
          }
        }
      }

      // ---- Kt = Sminv * HP; store -Kt^T transposed (lane j owns column j of Kt) ----
      float hp[MDIM], kt[MDIM];
      {
        int j = (lane < SDIM) ? lane : 0;
#pragma unroll
        for (int mi = 0; mi < MDIM; ++mi) hp[mi] = HPs[mi * HPSTR + j];
#pragma unroll
        for (int mi = 0; mi < MDIM; ++mi) {
          float acc = 0.0f;
#pragma unroll
          for (int n = 0; n < MDIM; ++n) acc += I_[mi * 4 + n] * hp[n];
          kt[mi] = acc;
        }
        if (lane < SDIM) {
#pragma unroll
          for (int mi = 0; mi < MDIM; ++mi) nKt[lane * MDIM + mi] = -kt[mi];
        }
      }
      __syncthreads();

      // ---- downdate via WMMA (K=4, 4 ops): [P|m] += (-Kt^T) x [HP|mu] ----
      {
        v2f a0 = wmma_load_a(nKt, MDIM, 0, 0, lane);
        v2f a1 = wmma_load_a(nKt, MDIM, 16, 0, lane);
        v2f b0 = wmma_load_b(HPs, HPSTR, 0, 0, lane);
        v2f b1 = wmma_load_b(HPs, HPSTR, 0, 16, lane);
        v8f c00 = wmma_load_c(Pm, PSTR, 0, 0, lane);
        v8f c01 = wmma_load_c(Pm, PSTR, 0, 16, lane);
        v8f c10 = wmma_load_c(Pm, PSTR, 16, 0, lane);
        v8f c11 = wmma_load_c(Pm, PSTR, 16, 16, lane);
        c00 = WMMA_F32(a0, b0, c00);
        c01 = WMMA_F32(a0, b1, c01);
        c10 = WMMA_F32(a1, b0, c10);
        c11 = WMMA_F32(a1, b1, c11);
        wmma_store_d(Pm, PSTR, 0, 0, c00, lane);
        wmma_store_d(Pm, PSTR, 0, 16, c01, lane);
        wmma_store_d(Pm, PSTR, 16, 0, c10, lane);
        wmma_store_d(Pm, PSTR, 16, 16, c11, lane);
      }
      __syncthreads();

      // ---- finish mean update: col24 += Kt^T * y  (m_u = m - Kt^T*mu + Kt^T*y) ----
      if (lane < SDIM) {
        float add = kt[0] * Ys[0] + kt[1] * Ys[1] + kt[2] * Ys[2] + kt[3] * Ys[3];
        Pm[lane * PSTR + 24] += add;
      }
      __syncthreads();

      // ---- T1 = F * [P_u | m_u]  (24 WMMA; col 24 of Tm becomes predicted mean) ----
      {
        v8f c00 = {}, c01 = {}, c10 = {}, c11 = {};
#pragma unroll
        for (int k0 = 0; k0 < SDIM; k0 += 4) {
          v2f a0 = wmma_load_a(Fs, SDIM, 0, k0, lane);
          v2f a1 = wmma_load_a(Fs, SDIM, 16, k0, lane);
          v2f b0 = wmma_load_b(Pm, PSTR, k0, 0, lane);
          v2f b1 = wmma_load_b(Pm, PSTR, k0, 16, lane);
          c00 = WMMA_F32(a0, b0, c00);
          c01 = WMMA_F32(a0, b1, c01);
          c10 = WMMA_F32(a1, b0, c10);
          c11 = WMMA_F32(a1, b1, c11);
        }
        wmma_store_d(Tm, PSTR, 0, 0, c00, lane);
        wmma_store_d(Tm, PSTR, 0, 16, c01, lane);
        wmma_store_d(Tm, PSTR, 16, 0, c10, lane);
        wmma_store_d(Tm, PSTR, 16, 16, c11, lane);
      }
      __syncthreads();

      // ---- harvest predicted mean; P = T1 * F^T + Q (24 WMMA; col 24 re-zeroed) ----
      mreg = Tm[((lane < SDIM) ? lane : 0) * PSTR + 24];
      {
        v8f c00 = {}, c01 = {}, c10 = {}, c11 = {};
#pragma unroll
        for (int k0 = 0; k0 < SDIM; k0 += 4) {
          v2f a0 = wmma_load_a(Tm, PSTR, 0, k0, lane);
          v2f a1 = wmma_load_a(Tm, PSTR, 16, k0, lane);
          v2f b0 = wmma_load_bT(Fs, SDIM, k0, 0, lane);
          v2f b1 = wmma_load_bT(Fs, SDIM, k0, 16, lane);
          c00 = WMMA_F32(a0, b0, c00);
          c01 = WMMA_F32(a0, b1, c01);
          c10 = WMMA_F32(a1, b0, c10);
          c11 = WMMA_F32(a1, b1, c11);
        }
        wmma_store_d_plus_q(Pm, PSTR, 0, 0, c00, Qs, lane);
        wmma_store_d_plus_q(Pm, PSTR, 0, 16, c01, Qs, lane);
        wmma_store_d_plus_q(Pm, PSTR, 16, 0, c10, Qs, lane);
        wmma_store_d_plus_q(Pm, PSTR, 16, 16, c11, Qs, lane);
      }
    }

    wait_async0();     // next step's staged inputs resident in LDS
    __syncthreads();
  }
}

extern "C" void kernel_launch(void* const* d_in, const int* in_sizes, int n_in,
                              void* d_out, int out_size, void* d_ws, size_t ws_size,
                              hipStream_t stream) {
  (void)in_sizes; (void)n_in; (void)out_size; (void)d_ws; (void)ws_size;
  const float* yin = (const float*)d_in[0];
  const float* Fin = (const float*)d_in[1];
  const float* Qin = (const float*)d_in[2];
  const float* Hin = (const float*)d_in[3];
  const float* Rin = (const float*)d_in[4];
  const float* m0 = (const float*)d_in[5];
  const float* P0 = (const float*)d_in[6];
  float* out = (float*)d_out;
  kalman_wmma_kernel<<<dim3(GDIM), dim3(32), 0, stream>>>(yin, Fin, Qin, Hin, Rin, m0, P0, out);
}